// ChamferLoss_45707041964670
// MI455X (gfx1250) — compile-verified
//
#include <hip/hip_runtime.h>
#include <hip/hip_bf16.h>

typedef __attribute__((ext_vector_type(2))) float v2f;
typedef __attribute__((ext_vector_type(8))) float v8f;
typedef __attribute__((ext_vector_type(4))) int   v4i;

// pointer-to-AS1(v4i) and pointer-to-AS3(v4i) for the async copy builtin
typedef __attribute__((address_space(1))) v4i* gv4i_p;
typedef __attribute__((address_space(3))) v4i* lv4i_p;

#define BSZ   2
#define HH    128
#define WW    128
#define NPTS  (HH * WW)          // 16384 points per cloud
#define CHUNK 512                 // targets staged in LDS per pipeline stage
#define IPC   (CHUNK / 256)       // async issues per chunk per thread (=2)
#define ROWS_PER_BLOCK 128        // 8 waves * 16 query rows
#define DEG2RAD 0.017453292519943295f

// Workspace layout (floats):
//   qa_tar  [B*N*4]  A-format query vec  [x, y, z, 1]
//   tb_tar  [B*N*4]  B-format target vec [-2x, -2y, -2z, ||p||^2]
//   qa_fake [B*N*4]
//   tb_fake [B*N*4]
//   acc     [1]
#define OFF_QA_TAR  0
#define OFF_TB_TAR  (BSZ * NPTS * 4)
#define OFF_QA_FAKE (2 * BSZ * NPTS * 4)
#define OFF_TB_FAKE (3 * BSZ * NPTS * 4)
#define OFF_ACC     (4 * BSZ * NPTS * 4)

// gfx1250 async global->LDS path (ASYNCcnt), guarded so either toolchain builds.
#if defined(__has_builtin)
#if __has_builtin(__builtin_amdgcn_global_load_async_to_lds_b128) && \
    __has_builtin(__builtin_amdgcn_s_wait_asynccnt)
#define USE_ASYNC_LDS 1
#endif
#endif

// ---------------------------------------------------------------------------
// Kernel 1: spherical reprojection -> point clouds, packed into WMMA-friendly
// A-format and B-format float4 arrays. Also zeroes the global accumulator.
// ---------------------------------------------------------------------------
__global__ void transform_kernel(const float* __restrict__ fake,
                                 const float* __restrict__ tar,
                                 const float* __restrict__ sh,
                                 const float* __restrict__ sw,
                                 float* __restrict__ ws) {
    int idx = blockIdx.x * blockDim.x + threadIdx.x;   // [0, B*N)
    if (idx == 0) ws[OFF_ACC] = 0.0f;                  // stream-ordered before chamfer
    if (idx >= BSZ * NPTS) return;

    int n = idx % NPTS;
    int p = n / WW;            // row -> pitch
    int q = n % WW;            // col -> yaw

    const float Wf = 1285.0f, Hf = 438.0f;
    const float FARO_V = 123.5f, FARO_H = 360.0f;
    const float cropH = 384.0f, cropW = 384.0f;
    float crop_w_rad = sw[0] / Wf * FARO_H;
    float crop_h_rad = sh[0] / Hf * FARO_V;
    float fv_crop = FARO_V * cropH / Hf;
    float fh_crop = FARO_H * cropW / Wf;

    float yaw   = (-fh_crop * (float)q / (float)WW + crop_w_rad) * DEG2RAD;
    float pitch = (-fv_crop * (float)p / (float)HH + crop_h_rad) * DEG2RAD;

    float sy = __sinf(yaw),   cy = __cosf(yaw);
    float sp = __sinf(pitch), cp = __cosf(pitch);
    float dx = sp * sy, dy = sp * cy, dz = cp;

    float dt = tar[idx];
    float df = fake[idx];

    float4* qa_tar  = (float4*)(ws + OFF_QA_TAR);
    float4* tb_tar  = (float4*)(ws + OFF_TB_TAR);
    float4* qa_fake = (float4*)(ws + OFF_QA_FAKE);
    float4* tb_fake = (float4*)(ws + OFF_TB_FAKE);

    {   // tar cloud
        float x = dt * dx, y = dt * dy, z = dt * dz;
        float n2 = x * x + y * y + z * z;
        qa_tar[idx] = make_float4(x, y, z, 1.0f);
        tb_tar[idx] = make_float4(-2.0f * x, -2.0f * y, -2.0f * z, n2);
    }
    {   // fake cloud
        float x = df * dx, y = df * dy, z = df * dz;
        float n2 = x * x + y * y + z * z;
        qa_fake[idx] = make_float4(x, y, z, 1.0f);
        tb_fake[idx] = make_float4(-2.0f * x, -2.0f * y, -2.0f * z, n2);
    }
}

// ---------------------------------------------------------------------------
// Kernel 2: WMMA chamfer with double-buffered (async) LDS target staging.
// grid = (N/128, B, 2 directions), block = 256 (8 wave32s).
// ---------------------------------------------------------------------------
__global__ void __launch_bounds__(256)
chamfer_kernel(const float* __restrict__ ws_qa_tar,
               const float* __restrict__ ws_tb_tar,
               const float* __restrict__ ws_qa_fake,
               const float* __restrict__ ws_tb_fake,
               float* __restrict__ acc) {
    const int b   = blockIdx.y;
    const int dir = blockIdx.z;

    // dir 0: queries = tar, targets = fake. dir 1: queries = fake, targets = tar.
    const float* qa  = (dir == 0) ? ws_qa_tar  : ws_qa_fake;   // A-format queries
    const float* tbq = (dir == 0) ? ws_tb_tar  : ws_tb_fake;   // query norms (.w)
    const float* tb  = (dir == 0) ? ws_tb_fake : ws_tb_tar;    // B-format targets
    qa  += (size_t)b * NPTS * 4;
    tbq += (size_t)b * NPTS * 4;
    tb  += (size_t)b * NPTS * 4;

    __shared__ float4 sB[2][CHUNK];

    const int tid     = threadIdx.x;
    const int lane    = tid & 31;
    const int wave    = tid >> 5;
    const int halfsel = lane >> 4;      // 0: K=0,1 ; 1: K=2,3
    const int l16     = lane & 15;

    // A fragment: row striped across VGPRs in-lane (ISA 16x4 f32 A layout).
    const int arow = blockIdx.x * ROWS_PER_BLOCK + wave * 16 + l16;
    v2f a = *(const v2f*)(qa + (size_t)arow * 4 + 2 * halfsel);

    v8f runmin;
#pragma unroll
    for (int i = 0; i < 8; ++i) runmin[i] = 3.0e38f;
    v8f czero = {};

    // --- staging helper: each thread moves IPC float4 elements of a chunk ---
    auto issue_chunk = [&](int c, int buf) {
#pragma unroll
        for (int j = 0; j < IPC; ++j) {
            const int e = tid + j * 256;                       // element in chunk
            const float* src = tb + (size_t)(c * CHUNK + e) * 4;
#ifdef USE_ASYNC_LDS
            __builtin_amdgcn_global_load_async_to_lds_b128(
                (gv4i_p)src, (lv4i_p)&sB[buf][e],
                /*offset=*/0, /*cpol=*/0);
#else
            sB[buf][e] = *(const float4*)src;
#endif
        }
    };

    const int nchunks = NPTS / CHUNK;
    issue_chunk(0, 0);                                         // prologue

    for (int c = 0; c < nchunks; ++c) {
        const int buf = c & 1;
        if (c + 1 < nchunks) {
            issue_chunk(c + 1, buf ^ 1);
#ifdef USE_ASYNC_LDS
            __builtin_amdgcn_s_wait_asynccnt(IPC);             // chunk c landed
#endif
        } else {
#ifdef USE_ASYNC_LDS
            __builtin_amdgcn_s_wait_asynccnt(0);
#endif
        }
        __syncthreads();                                       // all waves' data visible

        const float2* sBf2 = (const float2*)sB[buf];
#pragma unroll 8
        for (int t = 0; t < CHUNK / 16; ++t) {
            // B fragment: column striped across lanes (mirror of A layout).
            const int col = t * 16 + l16;
            float2 bf = sBf2[col * 2 + halfsel];
            v2f bb; bb.x = bf.x; bb.y = bf.y;

            // D(16x16) = A(16x4) * B(4x16): d_ij = -2 x_i . y_j + ||y_j||^2
            v8f d = __builtin_amdgcn_wmma_f32_16x16x4_f32(
                false, a, false, bb, (short)0, czero, false, false);
#pragma unroll
            for (int i = 0; i < 8; ++i) runmin[i] = fminf(runmin[i], d[i]);
        }
        __syncthreads();                                       // done reading buf
    }

    // Column-min across the 16 lanes of each half-wave (D cols live in lanes).
#pragma unroll
    for (int m = 1; m < 16; m <<= 1) {
#pragma unroll
        for (int i = 0; i < 8; ++i)
            runmin[i] = fminf(runmin[i], __shfl_xor(runmin[i], m, 32));
    }

    // VGPR i of half h holds row (wavebase + i + 8*h); lane l16==0 has the min.
    if (l16 == 0) {
        const int rowbase = blockIdx.x * ROWS_PER_BLOCK + wave * 16 + 8 * halfsel;
        float s = 0.0f;
#pragma unroll
        for (int i = 0; i < 8; ++i) {
            float x2 = tbq[(size_t)(rowbase + i) * 4 + 3];     // ||x_row||^2
            s += fmaxf(runmin[i] + x2, 0.0f);
        }
        atomicAdd(acc, s);
    }
}

// ---------------------------------------------------------------------------
// Kernel 3: scale to mean(dist1) + mean(dist2) and emit scalar.
// ---------------------------------------------------------------------------
__global__ void finalize_kernel(const float* __restrict__ acc,
                                float* __restrict__ out) {
    out[0] = acc[0] * (1.0f / (float)(BSZ * NPTS));
}

extern "C" void kernel_launch(void* const* d_in, const int* in_sizes, int n_in,
                              void* d_out, int out_size, void* d_ws, size_t ws_size,
                              hipStream_t stream) {
    const float* fake = (const float*)d_in[0];
    const float* tar  = (const float*)d_in[1];
    const float* sh   = (const float*)d_in[2];
    const float* sw   = (const float*)d_in[3];
    float* ws  = (float*)d_ws;
    float* out = (float*)d_out;

    // 1) Build packed point clouds (+ zero accumulator).
    transform_kernel<<<(BSZ * NPTS) / 256, 256, 0, stream>>>(fake, tar, sh, sw, ws);

    // 2) Chamfer: grid (query tiles, batch, direction).
    dim3 grid(NPTS / ROWS_PER_BLOCK, BSZ, 2);
    chamfer_kernel<<<grid, 256, 0, stream>>>(ws + OFF_QA_TAR, ws + OFF_TB_TAR,
                                             ws + OFF_QA_FAKE, ws + OFF_TB_FAKE,
                                             ws + OFF_ACC);

    // 3) Reduce to scalar mean.
    finalize_kernel<<<1, 1, 0, stream>>>(ws + OFF_ACC, out);
}